// PositionalEncoderModule_48455821033878
// MI455X (gfx1250) — compile-verified
//
#include <hip/hip_runtime.h>

// Reference (fixed shape): x: [8, 4096, 1024] fp32.
// out[b,s,d] = in[b,s,d] + pe[s,d]
//   pe[s,d] = sin(s * 10000^(-2d/1024)) for even d, cos(...) for odd d.
//
// Pure streaming op: 134 MB read + 134 MB write = 268 MB -> ~11.5 us floor
// at 23.3 TB/s HBM. Design:
//   - 1 thread = one (s, d..d+3) quad -> b128 non-temporal load/store,
//     fully coalesced in wave32 (32 lanes x 16 B = 512 B per access).
//   - Issue all 8 batch loads FIRST (8 outstanding LOADcnt ops), then compute
//     the 4 sin/cos PE values under load latency, reuse them across all 8
//     batch slices (8x fewer transcendentals), then add + NT store.
//   - libm sinf/cosf (not native v_sin_f32): matches jnp.sin on fp32 args;
//     its slow path (|x|>=32768) is branched around at runtime since max
//     argument is 4095.

#define PE_B 8
#define PE_S 4096
#define PE_D 1024

typedef float v4f __attribute__((ext_vector_type(4)));

__global__ __launch_bounds__(256) void
pe_add_kernel(const float* __restrict__ in, float* __restrict__ out) {
    const int tid = blockIdx.x * blockDim.x + threadIdx.x;   // 0 .. S*D/4-1
    const int d4  = tid & (PE_D / 4 - 1);                    // quad index in row
    const int s   = tid >> 8;                                // PE_D/4 == 256
    const int d0  = d4 << 2;                                 // first dim of quad

    const size_t row = (size_t)s * PE_D + (size_t)d0;        // 16B-aligned
    const size_t bs  = (size_t)PE_S * PE_D;                  // batch stride

    // Issue all 8 b128 loads up front; PE math below overlaps their latency.
    v4f v[PE_B];
#pragma unroll
    for (int b = 0; b < PE_B; ++b) {
        v[b] = __builtin_nontemporal_load(
            reinterpret_cast<const v4f*>(in + (size_t)b * bs + row));
    }

    // inv_freq(d) = 10000^(-2d/D) = exp2(d * c), c = -2*log2(10000)/1024
    const float c   = -0.025952563241307517f;
    const float pos = (float)s;

    const float f0 = exp2f(c * (float)(d0 + 0));
    const float f1 = exp2f(c * (float)(d0 + 1));
    const float f2 = exp2f(c * (float)(d0 + 2));
    const float f3 = exp2f(c * (float)(d0 + 3));

    v4f pe;
    pe.x = sinf(pos * f0);   // d0   even -> sin
    pe.y = cosf(pos * f1);   // d0+1 odd  -> cos
    pe.z = sinf(pos * f2);
    pe.w = cosf(pos * f3);

#pragma unroll
    for (int b = 0; b < PE_B; ++b) {
        v4f o = v[b] + pe;
        __builtin_nontemporal_store(
            o, reinterpret_cast<v4f*>(out + (size_t)b * bs + row));
    }
}

extern "C" void kernel_launch(void* const* d_in, const int* in_sizes, int n_in,
                              void* d_out, int out_size, void* d_ws, size_t ws_size,
                              hipStream_t stream) {
    (void)in_sizes; (void)n_in; (void)out_size; (void)d_ws; (void)ws_size;

    const float* in  = (const float*)d_in[0];
    float*       out = (float*)d_out;

    // One thread per (s, d-quad): S * D/4 = 4096 * 256 threads.
    const int threads = 256;                       // 8 wave32s per block
    const int total   = PE_S * (PE_D / 4);
    const int blocks  = total / threads;           // 4096 blocks

    pe_add_kernel<<<blocks, threads, 0, stream>>>(in, out);
}